// hierGCN_2138893713778
// MI455X (gfx1250) — compile-verified
//
#include <hip/hip_runtime.h>
#include <hip/hip_bf16.h>
#include <math.h>

// ---------------------------------------------------------------------------
// hierGCN (3x GAT conv + FC) for MI455X / gfx1250, wave32.
//   GEMMs       : V_WMMA_F32_16X16X4_F32, one wave per 16x16 C tile, K=64.
//   Edge phases : edge-parallel kernels with f32 global atomics (max/add).
//   Scatter     : one wave per edge, float2 per lane, 64 atomic f32 adds/edge.
// ---------------------------------------------------------------------------

typedef __attribute__((ext_vector_type(2))) float v2f;
typedef __attribute__((ext_vector_type(8))) float v8f;

#define N_FEAT 64
#define BETA_C 0.5f
#define NEG_SLOPE_C 0.2f

// ---------------- WMMA GEMM: C[M x 64] = A[M x 64] * B[64 x 64] (+ bias) ----
// Block = 128 threads = 4 waves; each wave owns one 16-wide N tile.
// M must be a multiple of 16 (50000 = 3125 * 16).
__global__ void __launch_bounds__(128)
gemm64_wmma_f32(const float* __restrict__ A, const float* __restrict__ B,
                const float* __restrict__ bias, float* __restrict__ C) {
  const int lane  = threadIdx.x & 31;
  const int wave  = threadIdx.x >> 5;        // 0..3 -> N tile
  const int m0    = blockIdx.x * 16;
  const int n0    = wave * 16;
  const int mrow  = lane & 15;               // A: M index held by this lane
  const int ncol  = lane & 15;               // B/C: N index held by this lane
  const int khalf = lane >> 4;               // 0: K {0,1}, 1: K {2,3} within step

  const float* Arow = A + (size_t)(m0 + mrow) * N_FEAT + 2 * khalf;
  const float* Bcol = B + (size_t)(2 * khalf) * N_FEAT + n0 + ncol;

  v8f acc = {};
#pragma unroll
  for (int kk = 0; kk < N_FEAT; kk += 4) {
    v2f a, bf;
    a.x  = Arow[kk + 0];
    a.y  = Arow[kk + 1];
    bf.x = Bcol[(size_t)kk * N_FEAT];
    bf.y = Bcol[(size_t)kk * N_FEAT + N_FEAT];
    acc = __builtin_amdgcn_wmma_f32_16x16x4_f32(
        /*neg_a=*/false, a, /*neg_b=*/false, bf,
        /*c_mod=*/(short)0, acc, /*reuse_a=*/false, /*reuse_b=*/false);
  }

  float bv = bias ? bias[n0 + ncol] : 0.0f;
#pragma unroll
  for (int r = 0; r < 8; ++r) {
    const int row = m0 + r + 8 * khalf;      // C layout: VGPR r -> M=r (+8 hi half)
    C[(size_t)row * N_FEAT + n0 + ncol] = acc[r] + bv;
  }
}

// ---------------- per-node attention scalars: as = h.a_src, ad = h.a_dst ----
__global__ void attn_scalars(const float* __restrict__ h,
                             const float* __restrict__ a_src,
                             const float* __restrict__ a_dst,
                             float* __restrict__ as_, float* __restrict__ ad_,
                             int N) {
  const int i = blockIdx.x * blockDim.x + threadIdx.x;
  if (i >= N) return;
  const float* hp = h + (size_t)i * N_FEAT;
  float s = 0.f, d = 0.f;
#pragma unroll
  for (int f = 0; f < N_FEAT; ++f) {
    const float hv = hp[f];
    s = fmaf(hv, a_src[f], s);
    d = fmaf(hv, a_dst[f], d);
  }
  as_[i] = s;
  ad_[i] = d;
}

// ---------------- init m=-inf, denom=0, agg=0 -------------------------------
__global__ void init_segments(float* __restrict__ m, float* __restrict__ denom,
                              float* __restrict__ agg, int N) {
  const int i = blockIdx.x * blockDim.x + threadIdx.x;
  const int NF = N * N_FEAT;
  if (i < NF) agg[i] = 0.0f;
  if (i < N) {
    m[i]     = -INFINITY;
    denom[i] = 0.0f;
  }
}

__device__ __forceinline__ void atomicMaxF32(float* addr, float val) {
  // IEEE-monotone trick: signed-int max for >=0, unsigned-int min for <0.
  if (val >= 0.0f)
    atomicMax((int*)addr, __float_as_int(val));
  else
    atomicMin((unsigned int*)addr, __float_as_uint(val));
}

// ---------------- edge pass 1: e = leaky_relu(as[src]+ad[dst]); seg-max -----
__global__ void edge_logits_max(const int* __restrict__ src,
                                const int* __restrict__ dst,
                                const float* __restrict__ as_,
                                const float* __restrict__ ad_,
                                float* __restrict__ ebuf,
                                float* __restrict__ m, int E) {
  const int i = blockIdx.x * blockDim.x + threadIdx.x;
  if (i >= E) return;
  const int d = dst[i];
  float e = as_[src[i]] + ad_[d];
  e = (e > 0.0f) ? e : NEG_SLOPE_C * e;
  ebuf[i] = e;
  atomicMaxF32(m + d, e);
}

// ---------------- edge pass 2: ex = exp(e - m[dst]); seg-sum ----------------
__global__ void edge_exp_sum(const int* __restrict__ dst,
                             const float* __restrict__ m,
                             float* __restrict__ ebuf,
                             float* __restrict__ denom, int E) {
  const int i = blockIdx.x * blockDim.x + threadIdx.x;
  if (i >= E) return;
  const int d  = dst[i];
  const float ex = __expf(ebuf[i] - m[d]);
  ebuf[i] = ex;
  atomicAdd(denom + d, ex);
}

// ---------------- edge pass 3: agg[dst] += (ex/denom[dst]) * h[src] ---------
// One wave32 per edge; each lane moves a float2 slice of the 64-float row.
__global__ void __launch_bounds__(256)
edge_scatter(const int* __restrict__ src, const int* __restrict__ dst,
             const float* __restrict__ ebuf, const float* __restrict__ denom,
             const float* __restrict__ h, float* __restrict__ agg, int E) {
  const int gtid = blockIdx.x * blockDim.x + threadIdx.x;
  const int eid  = gtid >> 5;
  const int lane = threadIdx.x & 31;
  if (eid >= E) return;

  const int s = src[eid];
  const int d = dst[eid];
  const float w = ebuf[eid] / denom[d];

  const float2* hp = (const float2*)(h + (size_t)s * N_FEAT);
  // prefetch the next edge's source row while this one is in flight
  if (eid + 1 < E) {
    const int sn = src[eid + 1];
    __builtin_prefetch(h + (size_t)sn * N_FEAT + 2 * lane, 0, 3);
  }
  const float2 hv = hp[lane];

  float* ap = agg + (size_t)d * N_FEAT + 2 * lane;
  atomicAdd(ap + 0, w * hv.x);
  atomicAdd(ap + 1, w * hv.y);
}

// ---------------- residual: xnext = BETA*(agg + b) + (1-BETA)*x0 ------------
__global__ void residual_mix(const float* __restrict__ agg,
                             const float* __restrict__ b,
                             const float* __restrict__ x0,
                             float* __restrict__ xnext, int NF) {
  const int i = blockIdx.x * blockDim.x + threadIdx.x;
  if (i >= NF) return;
  const int f = i & (N_FEAT - 1);
  xnext[i] = BETA_C * (agg[i] + b[f]) + (1.0f - BETA_C) * x0[i];
}

// ---------------------------------------------------------------------------
extern "C" void kernel_launch(void* const* d_in, const int* in_sizes, int n_in,
                              void* d_out, int out_size, void* d_ws, size_t ws_size,
                              hipStream_t stream) {
  const float* x     = (const float*)d_in[0];
  const int*   edge  = (const int*)d_in[1];   // [2, E], src then dst
  const float* W     = (const float*)d_in[2];
  const float* a_src = (const float*)d_in[3];
  const float* a_dst = (const float*)d_in[4];
  const float* b     = (const float*)d_in[5];
  const float* fc_w  = (const float*)d_in[6];
  const float* fc_b  = (const float*)d_in[7];

  const int N  = in_sizes[0] / N_FEAT;        // 50000
  const int E  = in_sizes[1] / 2;             // 1600000
  const int NF = N * N_FEAT;

  const int* src = edge;
  const int* dst = edge + E;

  // -------- workspace carve-out (fp32) --------
  float* ws   = (float*)d_ws;
  float* hbuf = ws;                 ws += (size_t)NF;   // h = x_layer @ W
  float* xcur = ws;                 ws += (size_t)NF;   // x1/x2/x3
  float* agg  = ws;                 ws += (size_t)NF;   // segment sums
  float* asb  = ws;                 ws += (size_t)N;
  float* adb  = ws;                 ws += (size_t)N;
  float* mbuf = ws;                 ws += (size_t)N;
  float* dnm  = ws;                 ws += (size_t)N;
  float* ebuf = ws;                 ws += (size_t)E;

  const int TPB        = 256;
  const int gemm_grid  = N / 16;                          // 3125 blocks x 128 thr
  const int node_grid  = (N + TPB - 1) / TPB;
  const int nf_grid    = (NF + TPB - 1) / TPB;
  const int edge_grid  = (E + TPB - 1) / TPB;
  const int scat_grid  = (E * 32 + TPB - 1) / TPB;        // one wave per edge

  for (int layer = 0; layer < 3; ++layer) {
    const float* xin = (layer == 0) ? x : xcur;

    gemm64_wmma_f32<<<gemm_grid, 128, 0, stream>>>(xin, W, nullptr, hbuf);
    attn_scalars<<<node_grid, TPB, 0, stream>>>(hbuf, a_src, a_dst, asb, adb, N);
    init_segments<<<nf_grid, TPB, 0, stream>>>(mbuf, dnm, agg, N);
    edge_logits_max<<<edge_grid, TPB, 0, stream>>>(src, dst, asb, adb, ebuf, mbuf, E);
    edge_exp_sum<<<edge_grid, TPB, 0, stream>>>(dst, mbuf, ebuf, dnm, E);
    edge_scatter<<<scat_grid, TPB, 0, stream>>>(src, dst, ebuf, dnm, hbuf, agg, E);
    residual_mix<<<nf_grid, TPB, 0, stream>>>(agg, b, x, xcur, NF);
  }

  // outputs: (out = x3 @ fc_w + fc_b, h = x3), concatenated flat
  float* out_p = (float*)d_out;
  float* h_p   = out_p + (size_t)NF;
  hipMemcpyAsync(h_p, xcur, (size_t)NF * sizeof(float),
                 hipMemcpyDeviceToDevice, stream);
  gemm64_wmma_f32<<<gemm_grid, 128, 0, stream>>>(xcur, fc_w, fc_b, out_p);
}